// LennardJonesCoulomb_79852031967360
// MI455X (gfx1250) — compile-verified
//
#include <hip/hip_runtime.h>

// CDNA5 / gfx1250, wave32. One V_WMMA_F32_16X16X4_F32 per 16x16 pair-tile
// computes -2*dot(xi,xj) + |xi|^2 ; the LJ/Coulomb VALU math co-executes.
// Per-pair path uses raw v_rsq_f32 only; all sqrt() hoisted out of the loop.

typedef float v2f __attribute__((ext_vector_type(2)));
typedef float v8f __attribute__((ext_vector_type(8)));

#define KE_CONST 332.0636f
#define TILE 16

__global__ void zero_out_kernel(float* __restrict__ out, int n) {
    int i = blockIdx.x * blockDim.x + threadIdx.x;
    if (i < n) out[i] = 0.0f;
}

template<int WAVES_PER_BLOCK, int JCHUNKS>
__global__ __launch_bounds__(32 * WAVES_PER_BLOCK)
void lj_coul_wmma_kernel(const float* __restrict__ pos,      // (N,3)
                         const float* __restrict__ charges,  // (N,1)
                         const float* __restrict__ ljp,      // (N,2) sigma,eps
                         const int*   __restrict__ frames,   // (N,)
                         const int*   __restrict__ mols,     // (N,)
                         float*       __restrict__ out,      // (F,)
                         int N)
{
    __shared__ float partial[WAVES_PER_BLOCK][TILE];

    const int lane   = threadIdx.x & 31;
    const int wave   = threadIdx.x >> 5;
    const int laneHi = lane >> 4;      // 0: lanes 0-15, 1: lanes 16-31
    const int n      = lane & 15;      // column within tile for this lane

    const int i0      = blockIdx.x * TILE;   // row tile base
    const int jchunk  = blockIdx.y;
    const int ntilesJ = N / TILE;

    // ---------------- Row side (fixed for this block) ----------------
    // A-matrix (16x4 f32) VGPR layout: lanes 0-15 hold K=0 (v0), K=1 (v1);
    // lanes 16-31 hold K=2 (v0), K=3 (v1), all for row M = lane & 15.
    // A = [-2x, -2y, -2z, |xi|^2], B = [x, y, z, 1]^T  ->  D = -2*xi.xj + |xi|^2.
    const int rA = i0 + n;
    const float pix = pos[3 * rA + 0];
    const float piy = pos[3 * rA + 1];
    const float piz = pos[3 * rA + 2];
    const float ni  = pix * pix + piy * piy + piz * piz;

    v2f a;
    a.x = laneHi ? (-2.0f * piz) : (-2.0f * pix);
    a.y = laneHi ? ni            : (-2.0f * piy);

    // Per-row scalars for the 8 rows this lane owns in the C/D layout:
    // row M = laneHi*8 + v  (VGPR v: lanes 0-15 -> M=v, lanes 16-31 -> M=v+8)
    // Pre-derived: KE*q, 0.5*sigma, sqrt(eps) -> no sqrt in the pair loop.
    float keqi[8], hsgi[8], sepi[8];
    int   fri[8], mli[8];
    #pragma unroll
    for (int v = 0; v < 8; ++v) {
        const int r = i0 + laneHi * 8 + v;
        keqi[v] = KE_CONST * charges[r];
        hsgi[v] = 0.5f * ljp[2 * r + 0];
        sepi[v] = __builtin_amdgcn_sqrtf(ljp[2 * r + 1]);
        fri[v]  = frames[r];
        mli[v]  = mols[r];
    }

    float acc[8];
    #pragma unroll
    for (int v = 0; v < 8; ++v) acc[v] = 0.0f;

    // ---------------- Column-tile loop ----------------
    const int jstride = JCHUNKS * WAVES_PER_BLOCK;
    #pragma unroll 2
    for (int jt = jchunk * WAVES_PER_BLOCK + wave; jt < ntilesJ; jt += jstride) {
        const int j = jt * TILE + n;

        // prefetch next tile's positions (gfx1250 global_prefetch_b8)
        if (jt + jstride < ntilesJ) {
            __builtin_prefetch(&pos[3 * (j + jstride * TILE)], 0, 1);
        }

        const float pjx = pos[3 * j + 0];
        const float pjy = pos[3 * j + 1];
        const float pjz = pos[3 * j + 2];
        const float nj  = pjx * pjx + pjy * pjy + pjz * pjz;
        const float qj  = charges[j];
        const float hsgj  = 0.5f * ljp[2 * j + 0];
        const float sepj4 = 4.0f * __builtin_amdgcn_sqrtf(ljp[2 * j + 1]);
        const int   frj = frames[j];
        const int   mlj = mols[j];

        // B-matrix (4x16 f32): v0 holds K=0 (lanes 0-15) / K=2 (lanes 16-31),
        // v1 holds K=1 / K=3, column N = lane & 15.
        v2f b;
        b.x = laneHi ? pjz  : pjx;
        b.y = laneHi ? 1.0f : pjy;

        v8f c = {};
        // 8 args: (neg_a, A, neg_b, B, c_mod, C, reuse_a, reuse_b)
        v8f d = __builtin_amdgcn_wmma_f32_16x16x4_f32(
            false, a, false, b, (short)0, c, false, false);

        #pragma unroll
        for (int v = 0; v < 8; ++v) {
            const bool msk = (fri[v] == frj) & (mli[v] != mlj);
            // d[v] = -2*dot + |xi|^2  ->  r2 = d[v] + |xj|^2
            float r2 = msk ? (d[v] + nj) : 1.0f;
            r2 = fmaxf(r2, 1e-6f);
            const float rinv  = __builtin_amdgcn_rsqf(r2);   // raw v_rsq_f32
            const float rinv2 = rinv * rinv;

            const float coul = keqi[v] * qj * rinv;          // KE*qi*qj/r

            const float sig  = hsgi[v] + hsgj;               // 0.5*(si+sj)
            const float sr2  = sig * sig * rinv2;            // (sig/r)^2
            const float sr6  = sr2 * sr2 * sr2;
            const float e4   = sepi[v] * sepj4;              // 4*sqrt(ei*ej)
            const float lj   = e4 * sr6 * (sr6 - 1.0f);      // 4e(sr12-sr6)

            acc[v] += msk ? (coul + lj) : 0.0f;
        }
    }

    // ---------------- Reduce: columns -> row sums ----------------
    // Sum acc[v] over the 16 lanes of each half (xor masks stay in-half).
    #pragma unroll
    for (int v = 0; v < 8; ++v) {
        float s = acc[v];
        s += __shfl_xor(s, 1, 32);
        s += __shfl_xor(s, 2, 32);
        s += __shfl_xor(s, 4, 32);
        s += __shfl_xor(s, 8, 32);
        if (n == 0) partial[wave][laneHi * 8 + v] = s;
    }
    __syncthreads();

    // Combine the block's waves; one atomicAdd per row into out[frame].
    if (threadIdx.x < TILE) {
        float s = 0.0f;
        #pragma unroll
        for (int w = 0; w < WAVES_PER_BLOCK; ++w) s += partial[w][threadIdx.x];
        const int fr = frames[i0 + threadIdx.x];
        atomicAdd(&out[fr], s);
    }
}

extern "C" void kernel_launch(void* const* d_in, const int* in_sizes, int n_in,
                              void* d_out, int out_size, void* d_ws, size_t ws_size,
                              hipStream_t stream) {
    const float* pos     = (const float*)d_in[0];
    const float* charges = (const float*)d_in[1];
    const float* ljp     = (const float*)d_in[2];
    const int*   frames  = (const int*)d_in[3];
    const int*   mols    = (const int*)d_in[4];
    float*       out     = (float*)d_out;

    const int N = in_sizes[3];   // frames is (N,)

    // Output is atomically accumulated -> zero it first (harness poisons it).
    zero_out_kernel<<<(out_size + 63) / 64, 64, 0, stream>>>(out, out_size);

    constexpr int WAVES = 4;   // 128 threads/block
    constexpr int JCH   = 8;   // column-chunk parallelism
    dim3 grid(N / TILE, JCH);
    lj_coul_wmma_kernel<WAVES, JCH><<<grid, 32 * WAVES, 0, stream>>>(
        pos, charges, ljp, frames, mols, out, N);
}